// Concat_68882685493783
// MI455X (gfx1250) — compile-verified
//
#include <hip/hip_runtime.h>
#include <hip/hip_bf16.h>

// CDNA5 / gfx1250, wave32. Matrix math via V_WMMA_F32_16X16X4_F32.
// out0 stages activations into LDS with GLOBAL_LOAD_ASYNC_TO_LDS_B128
// (ASYNCcnt-tracked, CDNA5 ISA) and reads A fragments
// from LDS via ds_load_b64.

typedef __attribute__((ext_vector_type(2))) float v2f;
typedef __attribute__((ext_vector_type(8))) float v8f;

__device__ __forceinline__ v8f wmma_f32(v2f a, v2f b, v8f c) {
  // 8 args: (neg_a, A, neg_b, B, c_mod, C, reuse_a, reuse_b)
  return __builtin_amdgcn_wmma_f32_16x16x4_f32(false, a, false, b, (short)0, c,
                                               false, false);
}

// Async 16-byte global->LDS copy (per-lane). lds_off = LDS byte address
// (low 32 bits of a generic shared pointer), gaddr = global source.
__device__ __forceinline__ void async_ld16(unsigned lds_off, const float* gaddr) {
  asm volatile("global_load_async_to_lds_b128 %0, %1, off"
               :: "v"(lds_off), "v"(gaddr)
               : "memory");
}
__device__ __forceinline__ void wait_async0() {
  asm volatile("s_wait_asynccnt 0" ::: "memory");
}

#define ROW_F 480           // floats per input/output row
#define MUL0 128
#define MUL1 64
#define MUL2 32
#define PADK 260            // 256 K-values + 4 pad floats (bank-conflict-free)

// ---------------------------------------------------------------------------
// out0 = [s1|s2] @ W0 * inv0 + b0        (K=256, out cols 0..127)
// Block: 256 thr = 8 waves, 64 rows. Async-stage s1|s2 (64 x 256 f32) into
// LDS, then wave w computes col tile w over 4 row tiles (4 accumulators,
// one B fragment feeds 4 WMMAs). A fragments come from LDS (ds_load_b64).
// ---------------------------------------------------------------------------
__global__ __launch_bounds__(256) void out0_kernel(
    const float* __restrict__ x1, const float* __restrict__ x2,
    const float* __restrict__ W0, const float* __restrict__ b0,
    float* __restrict__ out, int N) {
  __shared__ float sA[64 * PADK];           // ~65 KB

  const int t    = threadIdx.x;
  const int lane = t & 31;
  const int wave = t >> 5;                  // col tile 0..7
  const int lrow = lane & 15;
  const int half = lane >> 4;               // 0 -> K{0,1}, 1 -> K{2,3}
  const int col  = wave * 16 + lrow;
  const int row0 = blockIdx.x * 64;
  if (row0 >= N) return;                    // wave-uniform (never for N=200000)

  // ---- async stage: 64 rows x 512B from x1 (K 0..127) and x2 (K 128..255).
  // One 16B chunk per thread per iteration; a wave covers one contiguous
  // 512B row segment -> fully coalesced.
#pragma unroll
  for (int it = 0; it < 8; ++it) {
    const int c  = t + it * 256;            // chunk id 0..2047
    const int r  = c >> 5;                  // source row 0..63
    const int ck = c & 31;                  // 16B chunk within 512B segment
    const float* g1 = x1 + (size_t)(row0 + r) * ROW_F + ck * 4;
    const float* g2 = x2 + (size_t)(row0 + r) * ROW_F + ck * 4;
    const unsigned l1 =
        (unsigned)(unsigned long long)&sA[r * PADK + ck * 4];
    const unsigned l2 =
        (unsigned)(unsigned long long)&sA[r * PADK + 128 + ck * 4];
    async_ld16(l1, g1);
    async_ld16(l2, g2);
  }
  wait_async0();
  __syncthreads();

  // ---- compute: 4 row tiles per wave, shared B fragment per k-step
  v8f acc0 = {}, acc1 = {}, acc2 = {}, acc3 = {};
  const float* pb = W0 + (size_t)(2 * half) * MUL0 + col;
  const int abase = lrow * PADK + 2 * half;

#pragma unroll 2
  for (int k = 0; k < 256; k += 4) {
    v2f b;  b.x = pb[(size_t)k * MUL0]; b.y = pb[(size_t)k * MUL0 + MUL0];
    v2f a0; a0.x = sA[abase + 0 * 16 * PADK + k]; a0.y = sA[abase + 0 * 16 * PADK + k + 1];
    v2f a1; a1.x = sA[abase + 1 * 16 * PADK + k]; a1.y = sA[abase + 1 * 16 * PADK + k + 1];
    v2f a2; a2.x = sA[abase + 2 * 16 * PADK + k]; a2.y = sA[abase + 2 * 16 * PADK + k + 1];
    v2f a3; a3.x = sA[abase + 3 * 16 * PADK + k]; a3.y = sA[abase + 3 * 16 * PADK + k + 1];
    acc0 = wmma_f32(a0, b, acc0);
    acc1 = wmma_f32(a1, b, acc1);
    acc2 = wmma_f32(a2, b, acc2);
    acc3 = wmma_f32(a3, b, acc3);
  }

  const float inv0 = 0.0625f;               // 1/sqrt(256)
  const float bias = b0[col];
#pragma unroll
  for (int r = 0; r < 8; ++r) {             // C layout: VGPR r -> M = half*8 + r
    const int row = row0 + half * 8 + r;
    out[(size_t)(row +  0) * ROW_F + col] = acc0[r] * inv0 + bias;
    out[(size_t)(row + 16) * ROW_F + col] = acc1[r] * inv0 + bias;
    out[(size_t)(row + 32) * ROW_F + col] = acc2[r] * inv0 + bias;
    out[(size_t)(row + 48) * ROW_F + col] = acc3[r] * inv0 + bias;
  }
}

// ---------------------------------------------------------------------------
// out1: GEMM over 3N flattened rows r' = 3n+i, K=128, 64 out cols.
// A[r'][m] = x{1,2}[n][128 + 3*m' + i].  out[n*480 + 128 + o*3 + i].
// Block: 8 waves, 64 rows'. Wave w -> col tile (w&3), row tiles 2*(w>>2)+{0,1}
// (2 accumulators, B fragment reused x2).
// ---------------------------------------------------------------------------
__global__ __launch_bounds__(256) void out1_kernel(
    const float* __restrict__ x1, const float* __restrict__ x2,
    const float* __restrict__ W1, float* __restrict__ out, int N) {
  const int lane = threadIdx.x & 31;
  const int wave = threadIdx.x >> 5;
  const int ct   = wave & 3;                  // col tile 0..3
  const int t0   = (wave >> 2) * 2;           // row tiles t0, t0+1
  const int lrow = lane & 15;
  const int half = lane >> 4;
  const int col  = ct * 16 + lrow;
  const int rp0a = blockIdx.x * 64 + t0 * 16;
  const int rp0b = rp0a + 16;
  if (rp0a >= 3 * N) return;                  // wave-uniform

  const unsigned rpa = (unsigned)(rp0a + lrow);
  const unsigned na  = rpa / 3u, ia = rpa - 3u * na;
  const unsigned rpb = (unsigned)(rp0b + lrow);
  const unsigned nb  = rpb / 3u, ib = rpb - 3u * nb;

  const float* pa1a = x1 + (size_t)na * ROW_F + 128 + ia + 6 * half; // +3/K
  const float* pa2a = x2 + (size_t)na * ROW_F + 128 + ia + 6 * half;
  const float* pa1b = x1 + (size_t)nb * ROW_F + 128 + ib + 6 * half;
  const float* pa2b = x2 + (size_t)nb * ROW_F + 128 + ib + 6 * half;

  v8f accA = {}, accB = {};
  // phase 1: m = 0..63 from x1 (v1), W1 rows 0..63
  {
    const float* pb = W1 + (size_t)(2 * half) * MUL1 + col;
#pragma unroll 4
    for (int k = 0; k < 64; k += 4) {
      v2f b; b.x = pb[(size_t)k * MUL1]; b.y = pb[(size_t)k * MUL1 + MUL1];
      v2f a; a.x = pa1a[3 * k]; a.y = pa1a[3 * k + 3];
      v2f c; c.x = pa1b[3 * k]; c.y = pa1b[3 * k + 3];
      accA = wmma_f32(a, b, accA);
      accB = wmma_f32(c, b, accB);
    }
  }
  // phase 2: m = 64..127 from x2 (v2), W1 rows 64..127
  {
    const float* pb = W1 + (size_t)(64 + 2 * half) * MUL1 + col;
#pragma unroll 4
    for (int k = 0; k < 64; k += 4) {
      v2f b; b.x = pb[(size_t)k * MUL1]; b.y = pb[(size_t)k * MUL1 + MUL1];
      v2f a; a.x = pa2a[3 * k]; a.y = pa2a[3 * k + 3];
      v2f c; c.x = pa2b[3 * k]; c.y = pa2b[3 * k + 3];
      accA = wmma_f32(a, b, accA);
      accB = wmma_f32(c, b, accB);
    }
  }

  const float inv1 = 0.08838834764831845f;    // 1/sqrt(128)
#pragma unroll
  for (int r = 0; r < 8; ++r) {
    const unsigned rsa = (unsigned)(rp0a + half * 8 + r);
    const unsigned nsa = rsa / 3u, isa = rsa - 3u * nsa;
    out[(size_t)nsa * ROW_F + 128 + (unsigned)col * 3u + isa] = accA[r] * inv1;
    const unsigned rsb = (unsigned)(rp0b + half * 8 + r);
    const unsigned nsb = rsb / 3u, isb = rsb - 3u * nsb;
    out[(size_t)nsb * ROW_F + 128 + (unsigned)col * 3u + isb] = accB[r] * inv1;
  }
}

// ---------------------------------------------------------------------------
// out2: GEMM over 5N flattened rows r'' = 5n+i, K=64, 32 out cols.
// A[r''][m] = x{1,2}[n][320 + 5*m' + i].  out[n*480 + 320 + o*5 + i].
// Block: 8 waves; wave w -> col tile (w&1), row tile (w>>1). 64 rows''/block.
// ---------------------------------------------------------------------------
__global__ __launch_bounds__(256) void out2_kernel(
    const float* __restrict__ x1, const float* __restrict__ x2,
    const float* __restrict__ W2, float* __restrict__ out, int N) {
  const int lane = threadIdx.x & 31;
  const int wave = threadIdx.x >> 5;
  const int ct   = wave & 1;                  // col tile 0..1
  const int rt   = wave >> 1;                 // row tile 0..3
  const int lrow = lane & 15;
  const int half = lane >> 4;
  const int col  = ct * 16 + lrow;
  const int rp0  = blockIdx.x * 64 + rt * 16;
  if (rp0 >= 5 * N) return;                   // wave-uniform

  const unsigned rp = (unsigned)(rp0 + lrow);
  const unsigned n  = rp / 5u;
  const unsigned i  = rp - 5u * n;

  const float* pa1 = x1 + (size_t)n * ROW_F + 320 + i + 10 * half; // +5 per K
  const float* pa2 = x2 + (size_t)n * ROW_F + 320 + i + 10 * half;

  v8f acc = {};
  // phase 1: m = 0..31 from x1 (t1), W2 rows 0..31
  {
    const float* pb = W2 + (size_t)(2 * half) * MUL2 + col;
#pragma unroll 4
    for (int k = 0; k < 32; k += 4) {
      v2f a; a.x = pa1[5 * k]; a.y = pa1[5 * k + 5];
      v2f b; b.x = pb[(size_t)k * MUL2]; b.y = pb[(size_t)k * MUL2 + MUL2];
      acc = wmma_f32(a, b, acc);
    }
  }
  // phase 2: m = 32..63 from x2 (t2), W2 rows 32..63
  {
    const float* pb = W2 + (size_t)(32 + 2 * half) * MUL2 + col;
#pragma unroll 4
    for (int k = 0; k < 32; k += 4) {
      v2f a; a.x = pa2[5 * k]; a.y = pa2[5 * k + 5];
      v2f b; b.x = pb[(size_t)k * MUL2]; b.y = pb[(size_t)k * MUL2 + MUL2];
      acc = wmma_f32(a, b, acc);
    }
  }

  const float inv2 = 0.125f;                  // 1/sqrt(64)
#pragma unroll
  for (int r = 0; r < 8; ++r) {
    const unsigned rps = (unsigned)(rp0 + half * 8 + r);
    const unsigned ns  = rps / 5u;
    const unsigned is  = rps - 5u * ns;
    out[(size_t)ns * ROW_F + 320 + (unsigned)col * 5u + is] = acc[r] * inv2;
  }
}

extern "C" void kernel_launch(void* const* d_in, const int* in_sizes, int n_in,
                              void* d_out, int out_size, void* d_ws, size_t ws_size,
                              hipStream_t stream) {
  const float* x1 = (const float*)d_in[0];
  const float* x2 = (const float*)d_in[1];
  const float* W0 = (const float*)d_in[2];
  const float* W1 = (const float*)d_in[3];
  const float* W2 = (const float*)d_in[4];
  const float* b0 = (const float*)d_in[5];
  float* out = (float*)d_out;
  const int N = in_sizes[0] / ROW_F;          // 200000

  // out0: 64 rows / block (200000/64 = 3125)
  const int g0 = (N + 63) / 64;
  out0_kernel<<<g0, 256, 0, stream>>>(x1, x2, W0, b0, out, N);
  // out1: 64 flattened rows (3N = 600000 -> 9375 blocks)
  const int g1 = (3 * N + 63) / 64;
  out1_kernel<<<g1, 256, 0, stream>>>(x1, x2, W1, out, N);
  // out2: 64 flattened rows (5N = 1000000 -> 15625 blocks)
  const int g2 = (5 * N + 63) / 64;
  out2_kernel<<<g2, 256, 0, stream>>>(x1, x2, W2, out, N);
}